// DTWRepetitiveInterestTracker_91096256348504
// MI455X (gfx1250) — compile-verified
//
#include <hip/hip_runtime.h>
#include <hip/hip_bf16.h>
#include <math.h>
#include <stdint.h>

typedef __attribute__((ext_vector_type(2))) float v2f;
typedef __attribute__((ext_vector_type(8))) float v8f;
typedef __attribute__((ext_vector_type(4))) unsigned int u32x4;
typedef __attribute__((ext_vector_type(4))) int i32x4;
typedef __attribute__((ext_vector_type(8))) int i32x8;

#define NBATCH   65536
#define TQ       16
#define FQ       129
#define DDIM     64
#define CUT      64
#define BPB      64          // batches per block
#define NTHREADS 128         // 4 wave32
#define ROWPITCH 65          // 64 data dwords + 1 TDM pad dword (bank-conflict free)
#define BATPITCH (3 * ROWPITCH)   // 195 dwords per batch record in LDS

__global__ __launch_bounds__(NTHREADS)
void DTWRepetitiveInterestTracker_91096256348504_kernel(
        const float* __restrict__ fh,
        const float* __restrict__ W1,
        const float* __restrict__ b1,
        const float* __restrict__ gamma,
        const float* __restrict__ beta,
        const float* __restrict__ W2,
        const float* __restrict__ b2,
        float* __restrict__ out)
{
    __shared__ float sW1[DDIM * DDIM];        // 16 KB: W1, K-major
    __shared__ float sH[BPB * DDIM];          // 16 KB: h = latest @ W1
    __shared__ float sRaw[BPB * BATPITCH];    // ~48.8 KB: TDM-loaded window (padded rows)

    const int tid  = threadIdx.x;
    const int lane = tid & 31;
    const int wave = tid >> 5;
    const int blockBatch0 = blockIdx.x * BPB;

    // ---------------- Phase 0: TDM bulk fetch + W1 staging (overlapped) ----------------
    if (wave == 0) {
        // Tensor Data Mover: 3-D tile, X=64 contiguous floats, Y=3 rows (stride 129),
        // Z=64 batches (stride 2064). 1 pad dword after every 64 dwords -> row pitch 65.
        const uint64_t ga = (uint64_t)(uintptr_t)(fh + (size_t)blockBatch0 * (TQ * FQ) + 13 * FQ);
        const uint32_t lds_byte =
            (uint32_t)(uint64_t)(__attribute__((address_space(3))) float*)&sRaw[0];

        u32x4 g0 = { 1u,                                  // count=1 (valid), user D#
                     lds_byte,                            // lds_addr
                     (uint32_t)ga,                        // global_addr[31:0]
                     (uint32_t)((ga >> 32) & 0x1FFFFFFu)  // global_addr[56:32]
                         | (2u << 30) };                  // type=2 ("image")
        i32x8 g1 = { (int)((2u << 16)                     // data_size = 4 bytes
                         | (1u << 20)                     // pad_enable
                         | (5u << 22)                     // pad_interval: 64 dwords
                         | (0u << 25)),                   // pad_amount: 1 dword
                     (int)(129u << 16),                   // tensor_dim0[15:0] -> bits[63:48]
                     (int)(3u << 16),                     // dim0 hi=0 | tensor_dim1[15:0]
                     (int)(64u << 16),                    // dim1 hi=0 | tile_dim0=64
                     (int)((64u << 16) | 3u),             // tile_dim1=3 | tile_dim2=64
                     129,                                 // tensor_dim0_stride[31:0]
                     (int)(2064u << 16),                  // stride0 hi=0 | dim1_stride[15:0]
                     0 };                                 // tensor_dim1_stride[47:16]
        i32x4 g2 = { 64, 0, 0, 0 };                       // tensor_dim2 = 64
        i32x4 g3 = { 0, 0, 0, 0 };
#if __clang_major__ >= 23
        i32x8 g4 = { 0, 0, 0, 0, 0, 0, 0, 0 };
        __builtin_amdgcn_tensor_load_to_lds(g0, g1, g2, g3, g4, 0);
#else
        __builtin_amdgcn_tensor_load_to_lds(g0, g1, g2, g3, 0);
#endif
        __builtin_amdgcn_s_wait_tensorcnt(0);
    } else if (tid >= 64) {
        // waves 2-3: coalesced stage of W1 (64x64, K-major) into LDS, overlaps the DMA
        const int t = tid - 64;
        #pragma unroll 16
        for (int j = t; j < DDIM * DDIM; j += 64) sW1[j] = W1[j];
    }
    __syncthreads();

    // ---------------- Phase 1: per-batch stats, dtw, normalize row-15 in place ----------------
    if (tid < BPB) {
        const int gb = blockBatch0 + tid;
        float* rw = &sRaw[tid * BATPITCH];      // rows at +0, +65, +130
        float sum = 0.f, sumsq = 0.f, dtw = 0.f;
        #pragma unroll 8
        for (int j = 0; j < CUT; ++j) {
            float r13 = rw[j];
            float r14 = rw[ROWPITCH + j];
            float r15 = rw[2 * ROWPITCH + j];
            float a13 = fminf(fmaxf(fabsf(r13), 1e-5f), 10.f);
            float a14 = fminf(fmaxf(fabsf(r14), 1e-5f), 10.f);
            float a15 = fminf(fmaxf(fabsf(r15), 1e-5f), 10.f);
            sum   += a13 + a14 + a15;
            sumsq += a13 * a13 + a14 * a14 + a15 * a15;
            float d = r13 - r15;
            dtw += d * d;
        }
        const float n = 192.f;
        float mean = sum / n;
        float var  = (sumsq - sum * mean) / (n - 1.f);   // ddof=1
        float inv  = 1.f / (sqrtf(fmaxf(var, 0.f)) + 1e-5f);
        float* rw2 = rw + 2 * ROWPITCH;
        #pragma unroll 8
        for (int j = 0; j < CUT; ++j) {
            float v = fminf(fmaxf(fabsf(rw2[j]), 1e-5f), 10.f);
            rw2[j] = fminf(fmaxf((v - mean) * inv, -3.f), 3.f);  // WMMA A feed, in place
        }
        out[NBATCH + gb] = expf(-sqrtf(dtw) * (1.f / 256.f));
    }
    __syncthreads();

    // ---------------- Phase 2: h = latest @ W1 via V_WMMA_F32_16X16X4_F32 ----------------
    // Each wave: M=16 batch rows, N=64 (4 tiles), K=64 (16 steps of 4).
    const int half = lane >> 4;   // 0: lanes 0-15, 1: lanes 16-31
    const int lm   = lane & 15;
    const int rowBase = wave * 16;

    v8f acc0 = {}, acc1 = {}, acc2 = {}, acc3 = {};
    const float* aRow = &sRaw[(rowBase + lm) * BATPITCH + 2 * ROWPITCH];

    for (int k0 = 0; k0 < DDIM; k0 += 4) {
        // A (16x4 f32): lane l -> M = l%16 ; VGPR j -> K = k0 + j + 2*(l/16)
        const int ka = k0 + 2 * half;
        v2f a;
        a.x = aRow[ka];
        a.y = aRow[ka + 1];
        // B (4x16 f32): VGPR j -> K = k0 + j + 2*(l/16), N = 16*tile + l%16
        v2f bb0, bb1, bb2, bb3;
        bb0.x = sW1[ka * DDIM +  0 + lm];  bb0.y = sW1[(ka + 1) * DDIM +  0 + lm];
        bb1.x = sW1[ka * DDIM + 16 + lm];  bb1.y = sW1[(ka + 1) * DDIM + 16 + lm];
        bb2.x = sW1[ka * DDIM + 32 + lm];  bb2.y = sW1[(ka + 1) * DDIM + 32 + lm];
        bb3.x = sW1[ka * DDIM + 48 + lm];  bb3.y = sW1[(ka + 1) * DDIM + 48 + lm];

        acc0 = __builtin_amdgcn_wmma_f32_16x16x4_f32(false, a, false, bb0, (short)0, acc0, false, false);
        acc1 = __builtin_amdgcn_wmma_f32_16x16x4_f32(false, a, false, bb1, (short)0, acc1, false, false);
        acc2 = __builtin_amdgcn_wmma_f32_16x16x4_f32(false, a, false, bb2, (short)0, acc2, false, false);
        acc3 = __builtin_amdgcn_wmma_f32_16x16x4_f32(false, a, false, bb3, (short)0, acc3, false, false);
    }

    // Spill D: VGPR r holds rows r (lanes 0-15) and r+8 (lanes 16-31)
    #pragma unroll
    for (int r = 0; r < 8; ++r) {
        const int m = rowBase + r + 8 * half;
        sH[m * DDIM +  0 + lm] = acc0[r];
        sH[m * DDIM + 16 + lm] = acc1[r];
        sH[m * DDIM + 32 + lm] = acc2[r];
        sH[m * DDIM + 48 + lm] = acc3[r];
    }
    __syncthreads();

    // ---------------- Phase 3: bias + LayerNorm + GELU + W2 dot + sigmoid ----------------
    const int b  = tid >> 1;     // local batch 0..63
    const int hh = tid & 1;      // half of the 64 features
    const float* hr = &sH[b * DDIM + hh * 32];

    float vals[32];
    float s = 0.f, sq = 0.f;
    #pragma unroll
    for (int j = 0; j < 32; ++j) {
        float v = hr[j] + b1[hh * 32 + j];
        vals[j] = v;
        s  += v;
        sq += v * v;
    }
    s  += __shfl_xor(s, 1, 32);
    sq += __shfl_xor(sq, 1, 32);
    const float mu   = s * (1.f / 64.f);
    const float varp = sq * (1.f / 64.f) - mu * mu;   // population variance
    const float rstd = rsqrtf(varp + 1e-5f);

    float dot = 0.f;
    #pragma unroll
    for (int j = 0; j < 32; ++j) {
        const int ncol = hh * 32 + j;
        float x = (vals[j] - mu) * rstd * gamma[ncol] + beta[ncol];
        float g = 0.5f * x * (1.f + erff(x * 0.70710678118f));   // exact GELU
        dot += g * W2[ncol];
    }
    dot += __shfl_xor(dot, 1, 32);
    if (hh == 0) {
        const float z = dot + b2[0];
        out[blockBatch0 + b] = 1.f / (1.f + expf(-z));
    }
}

extern "C" void kernel_launch(void* const* d_in, const int* in_sizes, int n_in,
                              void* d_out, int out_size, void* d_ws, size_t ws_size,
                              hipStream_t stream) {
    (void)in_sizes; (void)n_in; (void)out_size; (void)d_ws; (void)ws_size;
    const float* fh    = (const float*)d_in[0];
    const float* W1    = (const float*)d_in[1];
    const float* b1    = (const float*)d_in[2];
    const float* gamma = (const float*)d_in[3];
    const float* beta  = (const float*)d_in[4];
    const float* W2    = (const float*)d_in[5];
    const float* b2    = (const float*)d_in[6];
    float* out = (float*)d_out;

    dim3 grid(NBATCH / BPB);   // 1024 blocks
    dim3 block(NTHREADS);      // 4 wave32
    DTWRepetitiveInterestTracker_91096256348504_kernel<<<grid, block, 0, stream>>>(
        fh, W1, b1, gamma, beta, W2, b2, out);
}